// SCCT_NGT_20461224198555
// MI455X (gfx1250) — compile-verified
//
#include <hip/hip_runtime.h>
#include <math.h>

typedef float v2f __attribute__((ext_vector_type(2)));
typedef float v8f __attribute__((ext_vector_type(8)));

#define NXP 64
#define NUM_BINS 64

__device__ __forceinline__ float fast_exp2(float x) {
#if __has_builtin(__builtin_amdgcn_exp2f)
    return __builtin_amdgcn_exp2f(x);      // v_exp_f32
#else
    return exp2f(x);
#endif
}

__device__ __forceinline__ float fast_rcp(float x) {
#if __has_builtin(__builtin_amdgcn_rcpf)
    return __builtin_amdgcn_rcpf(x);       // v_rcp_f32
#else
    return 1.0f / x;
#endif
}

__device__ __forceinline__ float fast_ln(float x) {
#if __has_builtin(__builtin_amdgcn_logf)
    return __builtin_amdgcn_logf(x) * 0.6931471805599453f;  // v_log_f32 * ln2
#else
    return logf(x);
#endif
}

// Branchless tanh: tanh(x) = sign(x)*(1-e)/(1+e), e = exp(-2|x|) = 2^(-2*log2e*|x|)
__device__ __forceinline__ float tanh_fast(float x) {
    const float ax = fabsf(x);
    const float e  = fast_exp2(-2.885390081777927f * ax);
    const float r  = (1.0f - e) * fast_rcp(1.0f + e);
    return copysignf(r, x);
}

__global__ void __launch_bounds__(32)
scct_ngt_kernel(const float* __restrict__ u0,
                const float* __restrict__ w1, const float* __restrict__ b1,
                const float* __restrict__ w2, const float* __restrict__ b2,
                const float* __restrict__ wc, const float* __restrict__ bc,
                const float* __restrict__ gamma_p, const int* __restrict__ nt_p,
                float* __restrict__ out)
{
    __shared__ float su[NXP];   // state u
    __shared__ float sg[NXP];   // grad_x feature
    __shared__ float sl[NXP];   // lap_x feature
    __shared__ float st[NXP];   // tree output
    __shared__ int   shist[NUM_BINS];

    const int  lane = threadIdx.x;     // 0..31, single wave32
    const int  m    = lane & 15;
    const bool hi   = (lane >= 16);

    const float DXv    = 1.0f / (NXP - 1);
    const float inv2dx = 1.0f / (2.0f * DXv);
    const float invdx2 = 1.0f / (DXv * DXv);
    const float DTv    = 1e-4f;

    const float gma = gamma_p[0];
    const int   Nt  = nt_p[0];
    const float bcv = bc[0];

    // A matrices = weights (16 hidden x K=4, K3 zero-padded), WMMA A layout:
    // lanes 0-15 (M=m): {K0,K1}; lanes 16-31: {K2,K3=0}. Loaded ONCE.
    v2f A1, A2;
    if (!hi) {
        A1.x = w1[m * 3 + 0]; A1.y = w1[m * 3 + 1];
        A2.x = w2[m * 3 + 0]; A2.y = w2[m * 3 + 1];
    } else {
        A1.x = w1[m * 3 + 2]; A1.y = 0.0f;
        A2.x = w2[m * 3 + 2]; A2.y = 0.0f;
    }

    // Bias folded into WMMA C operand: C[r] = b[n_h], n_h = r + 8*hi (uniform per half).
    // Head weights wc[n_h] become per-register constants.
    v8f c1, c2;
    float wc1v[8], wc2v[8];
    const int nb = hi ? 8 : 0;
    #pragma unroll
    for (int r = 0; r < 8; ++r) {
        c1[r]   = b1[nb + r];
        c2[r]   = b2[nb + r];
        wc1v[r] = wc[nb + r];          // layer-1 head
        wc2v[r] = wc[16 + nb + r];     // layer-2 head
    }

    // B = feat^T (4 x 16): lanes 0-15 -> {u, u_x}; lanes 16-31 -> {u_xx, 0}.
    // Branchless via pointer select for B.x.
    const float* pbx = hi ? sl : su;

    const int i0 = lane;               // each lane owns 2 grid points
    const int i1 = lane + 32;

    su[i0] = u0[i0];
    su[i1] = u0[i1];
    __syncthreads();

    for (int step = 0; step < Nt; ++step) {
        // ---- branchless stencil (clamped neighbors + select-zero at boundary) ----
        const float uc0 = su[i0];
        const float uc1 = su[i1];

        const int   im0 = (i0 > 0) ? i0 - 1 : 0;
        const float um0 = su[im0], up0 = su[i0 + 1];          // i0+1 <= 32, in range
        const bool  in0 = (i0 > 0);
        const float g0  = in0 ? (up0 - um0) * inv2dx : 0.0f;
        const float l0  = in0 ? (up0 - 2.0f * uc0 + um0) * invdx2 : 0.0f;

        const int   ip1 = (i1 < NXP - 1) ? i1 + 1 : NXP - 1;
        const float um1 = su[i1 - 1], up1 = su[ip1];          // i1-1 >= 31, in range
        const bool  in1 = (i1 < NXP - 1);
        const float g1  = in1 ? (up1 - um1) * inv2dx : 0.0f;
        const float l1  = in1 ? (up1 - 2.0f * uc1 + um1) * invdx2 : 0.0f;

        sg[i0] = g0; sl[i0] = l0;
        sg[i1] = g1; sl[i1] = l1;
        const float core0 = 0.8f * l0 + 0.5f * uc0 - uc0 * uc0 * uc0;
        const float core1 = 0.8f * l1 + 0.5f * uc1 - uc1 * uc1 * uc1;
        __syncthreads();

        // ---- tree net: D = W x feat^T, columns = grid points ----
        #pragma unroll
        for (int t = 0; t < 4; ++t) {
            const int col = 16 * t + m;
            v2f bmat;
            bmat.x = pbx[col];                    // lo: u[col], hi: u_xx[col]
            bmat.y = hi ? 0.0f : sg[col];         // lo: u_x[col], hi: K3 pad
            v8f d1 = __builtin_amdgcn_wmma_f32_16x16x4_f32(
                false, A1, false, bmat, (short)0, c1, false, false);
            v8f d2 = __builtin_amdgcn_wmma_f32_16x16x4_f32(
                false, A2, false, bmat, (short)0, c2, false, false);
            // In-lane contraction over hidden units (rows of D)
            float s = 0.0f;
            #pragma unroll
            for (int r = 0; r < 8; ++r) {
                s += tanh_fast(d1[r]) * wc1v[r];
                s += tanh_fast(d2[r]) * wc2v[r];
            }
            // merge hidden 0-7 (lanes 0-15) with hidden 8-15 (lanes 16-31)
            s += __shfl_xor(s, 16, 32);
            st[col] = s + bcv;   // lanes m and m+16 write bit-identical value
        }
        __syncthreads();

        // ---- Euler update + Dirichlet BCs ----
        float un0 = uc0 + DTv * (core0 + gma * st[i0]);
        float un1 = uc1 + DTv * (core1 + gma * st[i1]);
        if (i0 == 0)       un0 = 0.0f;
        if (i1 == NXP - 1) un1 = 0.0f;
        __syncthreads();
        su[i0] = un0;
        su[i1] = un1;
        __syncthreads();
    }

    // ---- outputs + SCCT stats (one-time) ----
    const float uf0 = su[i0];
    const float uf1 = su[i1];
    out[i0] = uf0;
    out[i1] = uf1;

    float ss = uf0 * uf0 + uf1 * uf1;
    ss += __shfl_xor(ss, 1, 32);  ss += __shfl_xor(ss, 2, 32);
    ss += __shfl_xor(ss, 4, 32);  ss += __shfl_xor(ss, 8, 32);
    ss += __shfl_xor(ss, 16, 32);
    const float phi2 = ss * (1.0f / 64.0f);

    const float v0 = fabsf(uf0), v1 = fabsf(uf1);
    float mx = fmaxf(v0, v1);
    mx = fmaxf(mx, __shfl_xor(mx, 1, 32));
    mx = fmaxf(mx, __shfl_xor(mx, 2, 32));
    mx = fmaxf(mx, __shfl_xor(mx, 4, 32));
    mx = fmaxf(mx, __shfl_xor(mx, 8, 32));
    mx = fmaxf(mx, __shfl_xor(mx, 16, 32));
    const float vmax  = mx;
    const float denom = fmaxf(vmax, 1e-12f);

    shist[i0] = 0;
    shist[i1] = 0;
    __syncthreads();
    int bin0 = (int)((v0 / denom) * (float)NUM_BINS);
    int bin1 = (int)((v1 / denom) * (float)NUM_BINS);
    bin0 = bin0 > NUM_BINS - 1 ? NUM_BINS - 1 : bin0;
    bin1 = bin1 > NUM_BINS - 1 ? NUM_BINS - 1 : bin1;
    atomicAdd(&shist[bin0], 1);
    atomicAdd(&shist[bin1], 1);
    __syncthreads();

    const float invN = 1.0f / (64.0f + 1e-12f);
    const float p0 = (float)shist[i0] * invN;
    const float p1 = (float)shist[i1] * invN;
    float h = -(p0 * fast_ln(p0 + 1e-12f)) - (p1 * fast_ln(p1 + 1e-12f));
    h += __shfl_xor(h, 1, 32);  h += __shfl_xor(h, 2, 32);
    h += __shfl_xor(h, 4, 32);  h += __shfl_xor(h, 8, 32);
    h += __shfl_xor(h, 16, 32);
    if (vmax < 1e-12f) h = 0.0f;

    if (lane == 0) {
        out[64] = phi2;
        out[65] = h;
    }
}

extern "C" void kernel_launch(void* const* d_in, const int* in_sizes, int n_in,
                              void* d_out, int out_size, void* d_ws, size_t ws_size,
                              hipStream_t stream) {
    (void)in_sizes; (void)n_in; (void)out_size; (void)d_ws; (void)ws_size;
    scct_ngt_kernel<<<1, 32, 0, stream>>>(
        (const float*)d_in[0],  // u0   [64]
        (const float*)d_in[1],  // w1   [16,3]
        (const float*)d_in[2],  // b1   [16]
        (const float*)d_in[3],  // w2   [16,3]
        (const float*)d_in[4],  // b2   [16]
        (const float*)d_in[5],  // wc   [1,32]
        (const float*)d_in[6],  // bc   [1]
        (const float*)d_in[7],  // gamma scalar
        (const int*)d_in[8],    // Nt scalar
        (float*)d_out);         // 66 floats: u[64], phi2, H
}